// Net_66279935312282
// MI455X (gfx1250) — compile-verified
//
#include <hip/hip_runtime.h>
#include <hip/hip_bf16.h>

// ---------------------------------------------------------------------------
// GCN forward for MI455X (gfx1250, wave32).
//   - BatchNorm folded into GEMM weights (per-feature affine).
//   - GEMMs use V_WMMA_F32_16X16X4_F32; A and B tiles staged in LDS
//     (X tile padded 128->132 floats/row for bank-conflict-free column reads,
//      W stored as scale-folded K-pairs so each B fragment is one ds_load_b64).
//   - Out-of-range rows handled by clamp-on-load / mask-on-store (row M of C
//     depends only on row M of A, so no zero-fill or exec masking needed).
//   - Edge scatter: 1 wave / edge, float4 gather + global_atomic_add_f32.
// ---------------------------------------------------------------------------

typedef __attribute__((ext_vector_type(2))) float v2f;
typedef __attribute__((ext_vector_type(8))) float v8f;

#define N_NODES 50000
#define N_EDGES 800000
#define DIM 128
#define N_GRAPHS 512
#define N_CLASS 10
#define BN_EPS 1e-5f

#define XS_PITCH 132                 // 128 + 4 pad: 528B row stride (16B aligned,
                                     // lanes hit distinct LDS banks on col reads)
#define GEMM_LDS_FLOATS (128 * XS_PITCH + 128 * 64 + 64)

// ---------------------------------------------------------------------------
// Deterministic BN statistics: stage 1, per-block partial sums over rows.
// ---------------------------------------------------------------------------
__global__ __launch_bounds__(128) void k_stats_partial(
    const float* __restrict__ X, int nrows, int rowsPerBlock,
    float* __restrict__ psum, float* __restrict__ psumsq)
{
    int f  = threadIdx.x;
    int r0 = blockIdx.x * rowsPerBlock;
    float s = 0.f, sq = 0.f;
    for (int i = 0; i < rowsPerBlock; ++i) {
        int r = r0 + i;
        if (r < nrows) {
            float v = X[(size_t)r * DIM + f];
            s += v; sq += v * v;
        }
    }
    psum  [blockIdx.x * DIM + f] = s;
    psumsq[blockIdx.x * DIM + f] = sq;
}

// Stage 2: fold partials into fused-affine scale/shift:
//   scale[f] = gamma*rsqrt(var+eps);  shift[f] = beta - mean*scale
__global__ __launch_bounds__(128) void k_stats_final(
    const float* __restrict__ psum, const float* __restrict__ psumsq, int nb,
    float invn, const float* __restrict__ gamma, const float* __restrict__ beta,
    float* __restrict__ ss)
{
    int f = threadIdx.x;
    float s = 0.f, sq = 0.f;
    for (int b = 0; b < nb; ++b) {
        s  += psum  [b * DIM + f];
        sq += psumsq[b * DIM + f];
    }
    float m   = s * invn;
    float var = sq * invn - m * m;
    float sc  = rsqrtf(var + BN_EPS) * gamma[f];
    ss[f]       = sc;
    ss[DIM + f] = beta[f] - m * sc;
}

// ---------------------------------------------------------------------------
// Fused BN + GEMM:  Y = relu?((x*s+t) @ W + bias)
// Rewritten as     Y = x @ (s∘W)  +  (tᵀW + bias)   so BN is free at WMMA time.
// Block = 256 thr (8 waves): 128 rows x 64 cols (blockIdx.y selects col half).
// Dynamic LDS: Xs[128][132], Wp[64][64][2] (K-pairs), cvec[64].
// Inner loop per K-step: 1 ds_load_b64 (A) + 4 ds_load_b64 (B) + 4 WMMA.
// ---------------------------------------------------------------------------
__global__ __launch_bounds__(256) void k_gemm_bn(
    const float* __restrict__ X, int nrows,
    const float* __restrict__ ss,      // [0..127]=scale, [128..255]=shift
    const float* __restrict__ W,       // [128][128] (k-major, as x @ W)
    const float* __restrict__ bias,    // nullable
    float* __restrict__ Y, int relu)
{
    extern __shared__ float smem[];
    float* Xs   = smem;                      // 128 * 132
    float* Wp   = Xs + 128 * XS_PITCH;       // 128 * 64 (pair layout)
    float* cvec = Wp + 128 * 64;             // 64

    int t = threadIdx.x;
    int colBase = blockIdx.y * 64;
    int rowBase = blockIdx.x * 128;

    // Stage X tile: coalesced float4 global loads, clamp out-of-range rows.
    for (int i = t; i < 128 * 32; i += 256) {
        int row = i >> 5, c4 = (i & 31) * 4;
        int gr  = rowBase + row;
        if (gr >= nrows) gr = nrows - 1;          // clamp: garbage rows are
        float4 v = *(const float4*)&X[(size_t)gr * DIM + c4];  // masked at store
        *(float4*)&Xs[row * XS_PITCH + c4] = v;
    }
    // Stage scale-folded W as K-pairs: Wp[(k>>1)*128 + o*2 + (k&1)]
    for (int i = t; i < 128 * 64; i += 256) {
        int k = i >> 6, o = i & 63;
        Wp[((k >> 1) * 64 + o) * 2 + (k & 1)] = W[k * DIM + colBase + o] * ss[k];
    }
    // Per-column constant: shift contribution + bias.
    if (t < 64) {
        float acc = bias ? bias[colBase + t] : 0.f;
        for (int k = 0; k < DIM; ++k)
            acc += ss[DIM + k] * W[k * DIM + colBase + t];
        cvec[t] = acc;
    }
    __syncthreads();

    int wave  = t >> 5;
    int lane  = t & 31;
    int laneM = lane & 15;
    int khalf = lane >> 4;               // 0: K pair {k0,k0+1}; 1: {k0+2,k0+3}
    int waveRow = wave * 16;

    const float* aRow = &Xs[(waveRow + laneM) * XS_PITCH];

    v8f z = {};
    v8f acc0 = z, acc1 = z, acc2 = z, acc3 = z;

    for (int k0 = 0; k0 < DIM; k0 += 4) {
        int ka = k0 + khalf * 2;
        v2f a = *(const v2f*)&aRow[ka];                       // ds_load_b64
        int p = ka >> 1;
        v2f b0 = *(const v2f*)&Wp[(p * 64 +  0 + laneM) * 2]; // ds_load_b64
        acc0 = __builtin_amdgcn_wmma_f32_16x16x4_f32(false, a, false, b0, (short)0, acc0, false, false);
        v2f b1 = *(const v2f*)&Wp[(p * 64 + 16 + laneM) * 2];
        acc1 = __builtin_amdgcn_wmma_f32_16x16x4_f32(false, a, false, b1, (short)0, acc1, false, false);
        v2f b2 = *(const v2f*)&Wp[(p * 64 + 32 + laneM) * 2];
        acc2 = __builtin_amdgcn_wmma_f32_16x16x4_f32(false, a, false, b2, (short)0, acc2, false, false);
        v2f b3 = *(const v2f*)&Wp[(p * 64 + 48 + laneM) * 2];
        acc3 = __builtin_amdgcn_wmma_f32_16x16x4_f32(false, a, false, b3, (short)0, acc3, false, false);
    }

    // Epilogue: C/D layout — VGPR j: lanes 0-15 -> M=j, lanes 16-31 -> M=j+8.
    int mOff = khalf * 8;
    v8f accs[4] = {acc0, acc1, acc2, acc3};
    for (int nt = 0; nt < 4; ++nt) {
        int col = colBase + nt * 16 + laneM;
        float cadd = cvec[nt * 16 + laneM];
        for (int j = 0; j < 8; ++j) {
            int r = rowBase + waveRow + mOff + j;
            if (r < nrows) {
                float v = accs[nt][j] + cadd;
                if (relu) v = fmaxf(v, 0.f);
                Y[(size_t)r * DIM + col] = v;
            }
        }
    }
}

// ---------------------------------------------------------------------------
// Degree / normalization
// ---------------------------------------------------------------------------
__global__ void k_deg_init(float* __restrict__ deg, int n) {
    int i = blockIdx.x * blockDim.x + threadIdx.x;
    if (i < n) deg[i] = 1.0f;                      // self loop
}
__global__ void k_deg_edges(const int* __restrict__ col, float* __restrict__ deg, int nE) {
    int e = blockIdx.x * blockDim.x + threadIdx.x;
    if (e < nE) atomicAdd(&deg[col[e]], 1.0f);
}
__global__ void k_dinv(const float* __restrict__ deg, float* __restrict__ dinv, int n) {
    int i = blockIdx.x * blockDim.x + threadIdx.x;
    if (i < n) dinv[i] = rsqrtf(deg[i]);           // deg >= 1 (self loop)
}

// agg[i,f] = bias[f] + dinv[i]^2 * y[i,f]   (self-loop term + conv bias)
__global__ void k_agg_init(const float* __restrict__ Y, const float* __restrict__ dinv,
                           const float* __restrict__ bias, float* __restrict__ agg, int n) {
    int idx = blockIdx.x * blockDim.x + threadIdx.x;
    if (idx < n * DIM) {
        int node = idx >> 7, f = idx & (DIM - 1);
        float d = dinv[node];
        agg[idx] = bias[f] + d * d * Y[idx];
    }
}

// One wave per edge: lane loads float4 (16B/lane, 512B/wave coalesced),
// 4 atomic f32 adds into the destination row.
__global__ __launch_bounds__(256) void k_scatter(
    const float* __restrict__ Y, const int* __restrict__ row,
    const int* __restrict__ col, const float* __restrict__ dinv,
    float* __restrict__ agg, int nE)
{
    int wid  = (blockIdx.x * blockDim.x + threadIdx.x) >> 5;
    int lane = threadIdx.x & 31;
    if (wid >= nE) return;
    int r = row[wid], c = col[wid];
    float nrm = dinv[r] * dinv[c];
    float4 v = ((const float4*)(Y + (size_t)r * DIM))[lane];
    float* dst = agg + (size_t)c * DIM + lane * 4;
    atomicAdd(dst + 0, nrm * v.x);
    atomicAdd(dst + 1, nrm * v.y);
    atomicAdd(dst + 2, nrm * v.z);
    atomicAdd(dst + 3, nrm * v.w);
}

// x += relu(agg)   (residual)
__global__ void k_residual(float* __restrict__ X, const float* __restrict__ agg, int n) {
    int idx = blockIdx.x * blockDim.x + threadIdx.x;
    if (idx < n) X[idx] += fmaxf(agg[idx], 0.f);
}

__global__ void k_zero(float* __restrict__ p, int n) {
    int idx = blockIdx.x * blockDim.x + threadIdx.x;
    if (idx < n) p[idx] = 0.f;
}

// global_add_pool: h[batch[n], f] += x[n, f]
__global__ void k_pool(const float* __restrict__ X, const int* __restrict__ batch,
                       float* __restrict__ h, int n) {
    int idx = blockIdx.x * blockDim.x + threadIdx.x;
    if (idx < n * DIM) {
        int node = idx >> 7, f = idx & (DIM - 1);
        atomicAdd(&h[(size_t)batch[node] * DIM + f], X[idx]);
    }
}

// Final BN + 128->10 classifier + log_softmax. One block (128 thr) per graph.
__global__ __launch_bounds__(128) void k_classifier(
    const float* __restrict__ H1, const float* __restrict__ ss,
    const float* __restrict__ Wc, const float* __restrict__ bc,
    float* __restrict__ out)
{
    __shared__ float hrow[DIM];
    __shared__ float logit[N_CLASS];
    __shared__ float lse;
    int row = blockIdx.x, t = threadIdx.x;
    hrow[t] = H1[(size_t)row * DIM + t] * ss[t] + ss[DIM + t];
    __syncthreads();
    if (t < N_CLASS) {
        float acc = bc[t];
        for (int k = 0; k < DIM; ++k) acc += hrow[k] * Wc[k * N_CLASS + t];
        logit[t] = acc;
    }
    __syncthreads();
    if (t == 0) {
        float mx = logit[0];
        for (int c = 1; c < N_CLASS; ++c) mx = fmaxf(mx, logit[c]);
        float se = 0.f;
        for (int c = 0; c < N_CLASS; ++c) se += __expf(logit[c] - mx);
        lse = mx + __logf(se);
    }
    __syncthreads();
    if (t < N_CLASS) out[(size_t)row * N_CLASS + t] = logit[t] - lse;
}

// ---------------------------------------------------------------------------
// Host-side orchestration.
// Input order (setup_inputs dict insertion order, params flattened in order):
//  0:x  1:edge_index(int32)  2:batch(int32)
//  3..8:  bn_conv (g0,b0,g1,b1,g2,b2)
//  9..14: convs  (W0,b0,W1,b1,W2,b2)
//  15,16: bn_fc (g,b)   17,18: lins (W,b)
//  19,20: bn_hidden (g,b)   21,22: lin_class (W,b)
// ---------------------------------------------------------------------------
extern "C" void kernel_launch(void* const* d_in, const int* in_sizes, int n_in,
                              void* d_out, int out_size, void* d_ws, size_t ws_size,
                              hipStream_t stream) {
    (void)in_sizes; (void)n_in; (void)out_size; (void)ws_size;

    const int N = N_NODES, E = N_EDGES, G = N_GRAPHS;
    const size_t NH = (size_t)N * DIM;
    const int STATS_ROWS = 128;
    const int SB = (N + STATS_ROWS - 1) / STATS_ROWS;   // 391 partial blocks
    const size_t gemmLds = GEMM_LDS_FLOATS * sizeof(float);

    const float* x_in  = (const float*)d_in[0];
    const int*   ei    = (const int*)d_in[1];
    const int*   batch = (const int*)d_in[2];
    const int*   row   = ei;          // edge_index[0]
    const int*   col   = ei + E;      // edge_index[1]

    const float* bn_g[3] = {(const float*)d_in[3], (const float*)d_in[5], (const float*)d_in[7]};
    const float* bn_b[3] = {(const float*)d_in[4], (const float*)d_in[6], (const float*)d_in[8]};
    const float* cW[3]   = {(const float*)d_in[9], (const float*)d_in[11], (const float*)d_in[13]};
    const float* cb[3]   = {(const float*)d_in[10], (const float*)d_in[12], (const float*)d_in[14]};
    const float* fc_g = (const float*)d_in[15];
    const float* fc_b = (const float*)d_in[16];
    const float* linW = (const float*)d_in[17];
    const float* linb = (const float*)d_in[18];
    const float* bh_g = (const float*)d_in[19];
    const float* bh_b = (const float*)d_in[20];
    const float* clsW = (const float*)d_in[21];
    const float* clsb = (const float*)d_in[22];
    float* out = (float*)d_out;

    // Workspace carve-up (floats)
    float* w      = (float*)d_ws;
    float* xcur   = w;                 // N*128
    float* y      = xcur + NH;         // N*128
    float* agg    = y + NH;            // N*128
    float* psum   = agg + NH;          // SB*128
    float* psumsq = psum + (size_t)SB * DIM;
    float* ss     = psumsq + (size_t)SB * DIM;   // 256 (scale|shift)
    float* deg    = ss + 256;          // N
    float* dinv   = deg + N;           // N
    float* hpool  = dinv + N;          // G*128
    float* h1     = hpool + (size_t)G * DIM;     // G*128
    float* ss2    = h1 + (size_t)G * DIM;        // 256

    hipMemcpyAsync(xcur, x_in, NH * sizeof(float), hipMemcpyDeviceToDevice, stream);

    // Degrees (identical across layers — once)
    k_deg_init <<<(N + 255) / 256, 256, 0, stream>>>(deg, N);
    k_deg_edges<<<(E + 255) / 256, 256, 0, stream>>>(col, deg, E);
    k_dinv     <<<(N + 255) / 256, 256, 0, stream>>>(deg, dinv, N);

    const int rowBlocks = (N + 127) / 128;       // 391
    dim3 gemmGrid(rowBlocks, 2);

    for (int L = 0; L < 3; ++L) {
        k_stats_partial<<<SB, 128, 0, stream>>>(xcur, N, STATS_ROWS, psum, psumsq);
        k_stats_final  <<<1, 128, 0, stream>>>(psum, psumsq, SB, 1.0f / N,
                                               bn_g[L], bn_b[L], ss);
        k_gemm_bn<<<gemmGrid, 256, gemmLds, stream>>>(xcur, N, ss, cW[L], nullptr, y, 0);
        k_agg_init<<<((int)NH + 255) / 256, 256, 0, stream>>>(y, dinv, cb[L], agg, N);
        k_scatter <<<(E + 7) / 8, 256, 0, stream>>>(y, row, col, dinv, agg, E);
        k_residual<<<((int)NH + 255) / 256, 256, 0, stream>>>(xcur, agg, (int)NH);
    }

    // global_add_pool
    k_zero<<<(G * DIM + 255) / 256, 256, 0, stream>>>(hpool, G * DIM);
    k_pool<<<((int)NH + 255) / 256, 256, 0, stream>>>(xcur, batch, hpool, N);

    // FC block: relu(BN(h) @ W + b)
    k_stats_partial<<<1, 128, 0, stream>>>(hpool, G, G, psum, psumsq);
    k_stats_final  <<<1, 128, 0, stream>>>(psum, psumsq, 1, 1.0f / G, fc_g, fc_b, ss);
    dim3 fcGrid((G + 127) / 128, 2);
    k_gemm_bn<<<fcGrid, 256, gemmLds, stream>>>(hpool, G, ss, linW, linb, h1, 1);

    // Final BN + classifier + log_softmax
    k_stats_partial<<<1, 128, 0, stream>>>(h1, G, G, psum, psumsq);
    k_stats_final  <<<1, 128, 0, stream>>>(psum, psumsq, 1, 1.0f / G, bh_g, bh_b, ss2);
    k_classifier<<<G, 128, 0, stream>>>(h1, ss2, clsW, clsb, out);
}